// Offset2D_73272142070107
// MI455X (gfx1250) — compile-verified
//
#include <hip/hip_runtime.h>

// Problem constants (from reference): B=16, C=256, H=W=128
#define NB   16
#define NC   256
#define NH   128
#define NW   128
#define NHW  16384          // NH*NW = 2^14
#define EPSF 1e-5f

typedef float v2f __attribute__((ext_vector_type(2)));
typedef float v8f __attribute__((ext_vector_type(8)));

// ---------------------------------------------------------------------------
// Kernel 0: zero the feature-output region (float4 stores) and init att_acc=EPS
// ---------------------------------------------------------------------------
__global__ __launch_bounds__(256) void k_init(float* __restrict__ out_feat,
                                              float* __restrict__ att_acc,
                                              long long n4_feat, long long n_acc) {
    long long i = (long long)blockIdx.x * 256 + threadIdx.x;
    if (i < n4_feat) {
        ((float4*)out_feat)[i] = make_float4(0.f, 0.f, 0.f, 0.f);
    }
    if (i < n_acc) att_acc[i] = EPSF;
}

// ---------------------------------------------------------------------------
// Kernel 1: 1x1 conv as WMMA GEMM + offset/attention/idx generation +
//           attention splat into att_acc.
// One wave32 per 16-pixel tile. A = 16x4 f32 (pixels x channels),
// B = 4x16 f32 (channels x padded outputs), K=256 in 64 WMMA steps.
// ---------------------------------------------------------------------------
__global__ __launch_bounds__(256) void k_conv_wmma(
    const float* __restrict__ x,        // [B, C, HW]
    const float* __restrict__ cw,       // [3, C]
    const float* __restrict__ cb,       // [3]
    float* __restrict__ out_off,        // [B, 2, HW]  (second part of d_out)
    float* __restrict__ att,            // [B, HW]
    int*   __restrict__ idx,            // [B, HW]
    float* __restrict__ att_acc)        // [B, HW] (pre-set to EPS)
{
    __shared__ float Bpad[NC * 16];     // conv_w padded 3->16 (zeros), 16 KB
    __shared__ float Dtile[8 * 16 * 16]; // per-wave 16x16 D spill, 8 KB

    const int tid = threadIdx.x;

    // Stage zero-padded B matrix: Bpad[c*16 + n] = (n<3) ? w[n,c] : 0
    {
        const int c = tid;              // blockDim == NC == 256
        float* row = &Bpad[c * 16];
        row[0] = cw[c];
        row[1] = cw[NC + c];
        row[2] = cw[2 * NC + c];
#pragma unroll
        for (int nn = 3; nn < 16; ++nn) row[nn] = 0.f;
    }
    __syncthreads();

    const int wave = tid >> 5;
    const int lane = tid & 31;
    const int tile = blockIdx.x * 8 + wave;  // 16384 tiles total
    const int b     = tile >> 10;            // 1024 tiles per batch
    const int pbase = (tile & 1023) << 4;    // 16 pixels per tile
    const int n  = lane & 15;                // N index (output col / pixel-in-tile)
    const int kh = (lane >> 4) << 1;         // upper half-wave handles K+2,K+3

    // A-matrix per-lane pointer: rows = channels, cols = pixels (contiguous)
    const float* xp = x + ((long long)b * NC + kh) * NHW + pbase + n;
    const float* bp = &Bpad[kh * 16 + n];

    v8f acc = {};
#pragma unroll 4
    for (int t = 0; t < 64; ++t) {
        v2f a;  a.x  = xp[0];  a.y  = xp[NHW];   // A[M=n][K=kh], A[n][kh+1]
        v2f bm; bm.x = bp[0];  bm.y = bp[16];    // B[K=kh][N=n], B[kh+1][n]
        acc = __builtin_amdgcn_wmma_f32_16x16x4_f32(
                  false, a, false, bm, (short)0, acc, false, false);
        xp += 4 * NHW;   // advance 4 channels
        bp += 64;        // 4 rows of 16
    }

    // Spill D (16x16) to LDS: lane L, elem r -> M = r + (L>=16 ? 8 : 0), N = L&15
    float* dt = &Dtile[wave * 256];
    const int mb = (lane >> 4) << 3;
#pragma unroll
    for (int r = 0; r < 8; ++r) dt[(mb + r) * 16 + n] = acc[r];
    __syncthreads();

    // Lanes 0..15: one pixel each (M = lane), finalize
    if (lane < 16) {
        const float oa0 = dt[lane * 16 + 0] + cb[0];
        const float oa1 = dt[lane * 16 + 1] + cb[1];
        const float oa2 = dt[lane * 16 + 2] + cb[2];
        const int p = pbase + lane;

        const float off_y = oa0 * (float)NH;
        const float off_x = oa1 * (float)NW;
        const float a = expf(oa2);

        const int h = p >> 7, w = p & (NW - 1);
        int iy = (int)rintf((float)h + off_y);   // v_rndne: half-to-even == jnp.round
        int ix = (int)rintf((float)w + off_x);
        iy = iy < 0 ? 0 : (iy > NH - 1 ? NH - 1 : iy);
        ix = ix < 0 ? 0 : (ix > NW - 1 ? NW - 1 : ix);
        const int di = (iy << 7) | ix;

        const long long bo = (long long)b * NHW;
        out_off[2 * bo + p]       = off_y;       // [B,2,HW] channel 0
        out_off[2 * bo + NHW + p] = off_x;       // channel 1
        att[bo + p] = a;
        idx[bo + p] = di;
        atomicAdd(&att_acc[bo + di], a);
    }
}

// ---------------------------------------------------------------------------
// Kernel 2: fold normalization into the source side:
//           att[p] <- att[p] / att_acc[idx[p]]   (in place)
// ---------------------------------------------------------------------------
__global__ __launch_bounds__(256) void k_scale(float* __restrict__ att,
                                               const int* __restrict__ idx,
                                               const float* __restrict__ att_acc) {
    const int i = blockIdx.x * 256 + threadIdx.x;   // < NB*NHW
    const int b = i >> 14;
    att[i] = att[i] / att_acc[(b << 14) + idx[i]];
}

// ---------------------------------------------------------------------------
// Kernel 3: scatter x * scale into out with fp32 global atomics.
// idx/scale are 2 MB total -> L2-resident across the whole C loop.
// ---------------------------------------------------------------------------
__global__ __launch_bounds__(256) void k_scatter(const float* __restrict__ x,
                                                 const float* __restrict__ scale,
                                                 const int*   __restrict__ idx,
                                                 float* __restrict__ out) {
    const long long i = (long long)blockIdx.x * 256 + threadIdx.x; // < B*C*HW
    const int p        = (int)(i & (NHW - 1));
    const long long bc = i >> 14;                 // b*NC + c
    const int bp       = (int)(((bc >> 8) << 14) + p); // b*NHW + p
    const float v = x[i] * scale[bp];
    const long long dest = (bc << 14) + idx[bp];
    atomicAdd(out + dest, v);
}

// ---------------------------------------------------------------------------
extern "C" void kernel_launch(void* const* d_in, const int* in_sizes, int n_in,
                              void* d_out, int out_size, void* d_ws, size_t ws_size,
                              hipStream_t stream) {
    const float* x  = (const float*)d_in[0];   // [16,256,128,128]
    const float* cw = (const float*)d_in[1];   // [3,256]
    const float* cb = (const float*)d_in[2];   // [3]

    float* out      = (float*)d_out;                        // [B,C,HW]
    float* out_off  = out + (long long)NB * NC * NHW;       // [B,2,HW]

    float* att      = (float*)d_ws;                         // B*HW f32
    int*   idx      = (int*)(att + (long long)NB * NHW);    // B*HW i32
    float* att_acc  = (float*)(idx + (long long)NB * NHW);  // B*HW f32

    const long long n_feat4 = ((long long)NB * NC * NHW) / 4;   // 16,777,216
    const long long n_acc   = (long long)NB * NHW;              // 262,144

    hipLaunchKernelGGL(k_init, dim3((unsigned)((n_feat4 + 255) / 256)), dim3(256),
                       0, stream, out, att_acc, n_feat4, n_acc);

    // 16384 tiles / 8 waves per block = 2048 blocks
    hipLaunchKernelGGL(k_conv_wmma, dim3(2048), dim3(256), 0, stream,
                       x, cw, cb, out_off, att, idx, att_acc);

    hipLaunchKernelGGL(k_scale, dim3((unsigned)(n_acc / 256)), dim3(256), 0, stream,
                       att, idx, att_acc);

    hipLaunchKernelGGL(k_scatter, dim3((unsigned)(((long long)NB * NC * NHW) / 256)),
                       dim3(256), 0, stream, x, att, idx, out);
}